// DAttention_76579266887926
// MI455X (gfx1250) — compile-verified
//
#include <hip/hip_runtime.h>
#include <hip/hip_bf16.h>

typedef float v2f __attribute__((ext_vector_type(2)));
typedef float v8f __attribute__((ext_vector_type(8)));

__device__ __forceinline__ v8f wmma_f32x4(v2f a, v2f b, v8f c) {
    // D = A(16x4) * B(4x16) + C(16x16), fp32, wave32
    return __builtin_amdgcn_wmma_f32_16x16x4_f32(false, a, false, b, (short)0, c, false, false);
}

// ---------------- Kernel 1: masked mean pool of queries ----------------
__global__ void pool_kernel(const float* __restrict__ q, const float* __restrict__ mask,
                            float* __restrict__ pooled) {
    int b = blockIdx.x;          // 0..7
    int c = threadIdx.x;         // 0..255
    const float* qb = q + (long long)b * 2048 * 256 + c;
    const float* mb = mask + (long long)b * 2048;
    float acc = 0.f, ms = 0.f;
    for (int l = 0; l < 2048; ++l) {
        float m = mb[l];
        acc += qb[(long long)l * 256] * m;
        ms  += m;
    }
    pooled[b * 256 + c] = acc / (ms + 1e-6f);
}

// ---------------- Kernel 2: style = pooled @ Wq^T + bq ----------------
__global__ void style_kernel(const float* __restrict__ pooled, const float* __restrict__ Wq,
                             const float* __restrict__ bq, float* __restrict__ style) {
    int b = blockIdx.x, o = threadIdx.x;
    const float* pb = pooled + b * 256;
    const float* wr = Wq + o * 256;
    float acc = bq[o];
    for (int i = 0; i < 256; ++i) acc += pb[i] * wr[i];
    style[b * 256 + o] = acc;
}

// ---------------- Kernel 3: modulate + demodulate weights ----------------
__global__ void wmod_kernel(const float* __restrict__ Wmod, const float* __restrict__ style,
                            float* __restrict__ wmod) {
    __shared__ float red[256];
    int o = blockIdx.x, b = blockIdx.y, i = threadIdx.x;
    float w = Wmod[o * 256 + i] * (style[b * 256 + i] + 1.0f);
    red[i] = w * w;
    __syncthreads();
    for (int s = 128; s > 0; s >>= 1) {
        if (i < s) red[i] += red[i + s];
        __syncthreads();
    }
    float d = rsqrtf(red[0] + 1e-8f);
    wmod[((long long)b * 256 + o) * 256 + i] = w * d;
}

// ---------------- Generic wave WMMA-f32 GEMM: C = A*B (+bias) ----------------
// A: MxK row-major (lda). B: KxN row-major (ldb), or BTRANS: elem(k,n) = B[n*ldb+k].
// One wave computes a 16 x (16*NT) tile. K multiple of 4, M mult of 16, N mult of 16*NT.
// BIAS: 0 none, 1 per-row (bias[m]), 2 per-col (bias[n]).
// CTRANS: 0 -> C[row*ldc+col], 1 -> C[col*ldc+row] (store transposed).
// Batch: A/B advance by sA/sB with power-of-two wrap masks aMask/bMask.
template <int BTRANS, int BIAS, int NT, int CTRANS>
__global__ __launch_bounds__(32)
void wmma_gemm_kernel(const float* __restrict__ A, long long sA, unsigned aMask, int lda,
                      const float* __restrict__ B, long long sB, unsigned bMask, int ldb,
                      const float* __restrict__ bias,
                      float* __restrict__ C, long long sC, int ldc, int K) {
    int lane = threadIdx.x;
    int half = lane >> 4;
    int l16  = lane & 15;
    int ks   = half * 2;
    int n0 = blockIdx.x * (16 * NT);
    int m0 = blockIdx.y * 16;
    unsigned z = blockIdx.z;
    A += (long long)(z & aMask) * sA;
    B += (long long)(z & bMask) * sB;
    C += (long long)z * sC;

    v8f acc[NT];
#pragma unroll
    for (int t = 0; t < NT; ++t) acc[t] = (v8f){0.f, 0.f, 0.f, 0.f, 0.f, 0.f, 0.f, 0.f};

    const float* arow = A + (long long)(m0 + l16) * lda + ks;
#pragma unroll 4
    for (int k = 0; k < K; k += 4) {
        v2f a;
        a.x = arow[k];
        a.y = arow[k + 1];
#pragma unroll
        for (int t = 0; t < NT; ++t) {
            v2f b;
            if (BTRANS) {
                const float* bp = B + (long long)(n0 + 16 * t + l16) * ldb + k + ks;
                b.x = bp[0];
                b.y = bp[1];
            } else {
                const float* bp = B + (long long)(k + ks) * ldb + (n0 + 16 * t + l16);
                b.x = bp[0];
                b.y = bp[ldb];
            }
            acc[t] = wmma_f32x4(a, b, acc[t]);
        }
    }
#pragma unroll
    for (int t = 0; t < NT; ++t) {
#pragma unroll
        for (int j = 0; j < 8; ++j) {
            int row = m0 + j + 8 * half;
            int col = n0 + 16 * t + l16;
            float vv = acc[t][j];
            if (BIAS == 1) vv += bias[row];
            else if (BIAS == 2) vv += bias[col];
            if (CTRANS) C[(long long)col * ldc + row] = vv;
            else        C[(long long)row * ldc + col] = vv;
        }
    }
}

// ---------------- Kernel 5: dwconv3x3 + LN + GELU + offset head + pos/ref ----------------
__global__ void offsets_kernel(const float* __restrict__ qt, const float* __restrict__ dw_w,
                               const float* __restrict__ dw_b, const float* __restrict__ ln_w,
                               const float* __restrict__ ln_b, const float* __restrict__ off_w,
                               float* __restrict__ pos_out, float* __restrict__ ref_out) {
    int idx = blockIdx.x * blockDim.x + threadIdx.x;   // 0..65535
    int bg = idx >> 10;
    int p  = idx & 1023;
    int iy = p >> 5, ix = p & 31;
    int b = bg >> 3, g = bg & 7;
    const float* base = qt + ((long long)b * 256 + g * 32) * 1024;

    float t[32];
    float mu = 0.f;
#pragma unroll
    for (int c = 0; c < 32; ++c) {
        float acc = dw_b[c];
        const float* ch = base + c * 1024;
        for (int ky = 0; ky < 3; ++ky) {
            int yy = iy + ky - 1;
            if (yy < 0 || yy > 31) continue;
            for (int kx = 0; kx < 3; ++kx) {
                int xx = ix + kx - 1;
                if (xx < 0 || xx > 31) continue;
                acc += ch[yy * 32 + xx] * dw_w[c * 9 + ky * 3 + kx];
            }
        }
        t[c] = acc;
        mu += acc;
    }
    mu *= (1.f / 32.f);
    float var = 0.f;
#pragma unroll
    for (int c = 0; c < 32; ++c) { float d = t[c] - mu; var += d * d; }
    var *= (1.f / 32.f);
    float rstd = rsqrtf(var + 1e-5f);

    float oy = 0.f, ox = 0.f;
#pragma unroll
    for (int c = 0; c < 32; ++c) {
        float xn = (t[c] - mu) * rstd * ln_w[c] + ln_b[c];
        float ge = 0.5f * xn * (1.0f + erff(xn * 0.7071067811865475f));
        oy += off_w[c] * ge;
        ox += off_w[32 + c] * ge;
    }
    const float orf = 2.0f / 31.0f;   // OFFSET_RANGE_FACTOR / (Hk-1)
    float ry = (iy + 0.5f) * (2.0f / 31.0f) - 1.0f;
    float rx = (ix + 0.5f) * (2.0f / 31.0f) - 1.0f;
    float py = tanhf(oy) * orf + ry;
    float px = tanhf(ox) * orf + rx;
    long long o = (long long)bg * 2048 + (long long)p * 2;
    pos_out[o] = py;  pos_out[o + 1] = px;
    ref_out[o] = ry;  ref_out[o + 1] = rx;
}

// ---------------- Kernel 6: bilinear grid sample (zeros pad, align_corners) ----------------
__global__ void gridsample_kernel(const float* __restrict__ x, const float* __restrict__ pos,
                                  float* __restrict__ xs) {
    int idx = blockIdx.x * blockDim.x + threadIdx.x;
    int bg = idx >> 10;
    int p  = idx & 1023;
    int b = bg >> 3, g = bg & 7;
    long long po = (long long)bg * 2048 + (long long)p * 2;
    float py = pos[po], px = pos[po + 1];
    float gx = (px + 1.f) * 0.5f * 31.f;
    float gy = (py + 1.f) * 0.5f * 31.f;
    float x0f = floorf(gx), y0f = floorf(gy);
    float wx1 = gx - x0f, wy1 = gy - y0f;
    int x0 = (int)x0f, y0 = (int)y0f;
    int x1 = x0 + 1, y1 = y0 + 1;
    float vx0 = (x0 >= 0 && x0 <= 31) ? 1.f : 0.f;
    float vx1 = (x1 >= 0 && x1 <= 31) ? 1.f : 0.f;
    float vy0 = (y0 >= 0 && y0 <= 31) ? 1.f : 0.f;
    float vy1 = (y1 >= 0 && y1 <= 31) ? 1.f : 0.f;
    int x0c = min(max(x0, 0), 31), x1c = min(max(x1, 0), 31);
    int y0c = min(max(y0, 0), 31), y1c = min(max(y1, 0), 31);
    float w00 = (1.f - wx1) * (1.f - wy1) * vx0 * vy0;
    float w10 = wx1 * (1.f - wy1) * vx1 * vy0;
    float w01 = (1.f - wx1) * wy1 * vx0 * vy1;
    float w11 = wx1 * wy1 * vx1 * vy1;
    const float* img = x + ((long long)(b & 1) * 256 + g * 32) * 1024;
#pragma unroll 4
    for (int c = 0; c < 32; ++c) {
        const float* ch = img + c * 1024;
        float val = w00 * ch[y0c * 32 + x0c] + w10 * ch[y0c * 32 + x1c]
                  + w01 * ch[y1c * 32 + x0c] + w11 * ch[y1c * 32 + x1c];
        xs[((long long)b * 256 + g * 32 + c) * 1024 + p] = val;
    }
}

// ---------------- Kernel 9: flash attention, one wave per 16-query tile ----------------
// kT: (8, 1024, 256)  key-major (transposed K projection output)
// v : (8, 256, 1024)  channel-major
__global__ __launch_bounds__(32)
void attn_kernel(const float* __restrict__ q, const float* __restrict__ kT,
                 const float* __restrict__ vbuf, float* __restrict__ attnout) {
    __shared__ float plds[16][17];
    int lane = threadIdx.x;
    int half = lane >> 4, l16 = lane & 15, ks = half * 2;
    int qt0  = blockIdx.x * 16;
    int head = blockIdx.y;                 // 0..63
    int b = head >> 3, h = head & 7;

    // Q A-fragments (16 queries x 32 channels): 8 k-chunks of 4
    v2f qa[8];
    {
        const float* qr = q + (((long long)b * 2048 + qt0 + l16) * 256) + h * 32;
#pragma unroll
        for (int kc = 0; kc < 8; ++kc) {
            qa[kc].x = qr[4 * kc + ks];
            qa[kc].y = qr[4 * kc + ks + 1];
        }
    }
    const float* kT_base = kT + (long long)b * 262144 + h * 32;              // + key*256 + ch
    const float* vb_base = vbuf + (long long)b * 262144 + (long long)(h * 32) * 1024;

    float runmax[8], runsum[8];
    v8f o0 = {0.f, 0.f, 0.f, 0.f, 0.f, 0.f, 0.f, 0.f};
    v8f o1 = {0.f, 0.f, 0.f, 0.f, 0.f, 0.f, 0.f, 0.f};
#pragma unroll
    for (int j = 0; j < 8; ++j) { runmax[j] = -3.0e38f; runsum[j] = 0.f; }
    const float scale = 0.17677669529663687f;   // 32^-0.5

    for (int kb = 0; kb < 64; ++kb) {
        // S = Q * K^T (16x16); K fragment rows are keys -> contiguous float2 per lane
        const float* krow = kT_base + (long long)(kb * 16 + l16) * 256 + ks;
        v8f s = {0.f, 0.f, 0.f, 0.f, 0.f, 0.f, 0.f, 0.f};
#pragma unroll
        for (int kc = 0; kc < 8; ++kc) {
            v2f bf;
            bf.x = krow[4 * kc];
            bf.y = krow[4 * kc + 1];
            s = wmma_f32x4(qa[kc], bf, s);
        }
        // online softmax per row (rows j+8*half, cols = 16 lanes of this half)
        float corr[8];
#pragma unroll
        for (int j = 0; j < 8; ++j) {
            float sv = s[j] * scale;
            float mx = sv;
            mx = fmaxf(mx, __shfl_xor(mx, 1, 16));
            mx = fmaxf(mx, __shfl_xor(mx, 2, 16));
            mx = fmaxf(mx, __shfl_xor(mx, 4, 16));
            mx = fmaxf(mx, __shfl_xor(mx, 8, 16));
            float nm = fmaxf(runmax[j], mx);
            float c  = expf(runmax[j] - nm);
            float pv = expf(sv - nm);
            float rs = pv;
            rs += __shfl_xor(rs, 1, 16);
            rs += __shfl_xor(rs, 2, 16);
            rs += __shfl_xor(rs, 4, 16);
            rs += __shfl_xor(rs, 8, 16);
            runsum[j] = runsum[j] * c + rs;
            runmax[j] = nm;
            corr[j] = c;
            plds[j + 8 * half][l16] = pv;
            o0[j] *= c;
            o1[j] *= c;
        }
        __syncthreads();
        // O += P * V^T  (16x32 as two 16x16 tiles)
#pragma unroll
        for (int kc2 = 0; kc2 < 4; ++kc2) {
            v2f pa;
            pa.x = plds[l16][4 * kc2 + ks];
            pa.y = plds[l16][4 * kc2 + ks + 1];
            const float* vp0 = vb_base + (long long)l16 * 1024 + kb * 16 + 4 * kc2 + ks;
            const float* vp1 = vb_base + (long long)(l16 + 16) * 1024 + kb * 16 + 4 * kc2 + ks;
            v2f b0; b0.x = vp0[0]; b0.y = vp0[1];
            v2f b1; b1.x = vp1[0]; b1.y = vp1[1];
            o0 = wmma_f32x4(pa, b0, o0);
            o1 = wmma_f32x4(pa, b1, o1);
        }
        __syncthreads();
    }
    // normalize + store: attnout[b][l][h*32 + ch]
#pragma unroll
    for (int j = 0; j < 8; ++j) {
        float inv = 1.0f / runsum[j];
        int l = qt0 + j + 8 * half;
        float* op = attnout + ((long long)b * 2048 + l) * 256 + h * 32;
        op[l16]      = o0[j] * inv;
        op[l16 + 16] = o1[j] * inv;
    }
}

extern "C" void kernel_launch(void* const* d_in, const int* in_sizes, int n_in,
                              void* d_out, int out_size, void* d_ws, size_t ws_size,
                              hipStream_t stream) {
    (void)in_sizes; (void)n_in; (void)out_size; (void)ws_size;
    const float* x    = (const float*)d_in[0];
    const float* q    = (const float*)d_in[1];
    const float* mask = (const float*)d_in[2];
    const float* Wq   = (const float*)d_in[3];
    const float* bq   = (const float*)d_in[4];
    const float* Wmod = (const float*)d_in[5];
    const float* dw_w = (const float*)d_in[6];
    const float* dw_b = (const float*)d_in[7];
    const float* ln_w = (const float*)d_in[8];
    const float* ln_b = (const float*)d_in[9];
    const float* off_w= (const float*)d_in[10];
    const float* Wk   = (const float*)d_in[11];
    const float* bk   = (const float*)d_in[12];
    const float* Wv   = (const float*)d_in[13];
    const float* bv   = (const float*)d_in[14];
    const float* Wo   = (const float*)d_in[15];
    const float* bo   = (const float*)d_in[16];

    float* out = (float*)d_out;
    float* y_out   = out;                       // (8,2048,256)
    float* pos_out = out + 4194304;             // (8,8,32,32,2)
    float* ref_out = out + 4194304 + 131072;    // (8,8,32,32,2)

    float* ws = (float*)d_ws;
    float* pooled  = ws;  ws += 2048;
    float* style   = ws;  ws += 2048;
    float* wmod    = ws;  ws += 524288;         // (8,256,256)
    float* qt      = ws;  ws += 2097152;        // (8,256,1024)
    float* xs      = ws;  ws += 2097152;        // (8,256,1024)
    float* kbufT   = ws;  ws += 2097152;        // (8,1024,256)  transposed K
    float* vbuf    = ws;  ws += 2097152;        // (8,256,1024)
    float* attnout = ws;  ws += 4194304;        // (8,2048,256)

    pool_kernel<<<8, 256, 0, stream>>>(q, mask, pooled);
    style_kernel<<<8, 256, 0, stream>>>(pooled, Wq, bq, style);
    wmod_kernel<<<dim3(256, 8), 256, 0, stream>>>(Wmod, style, wmod);

    // qt[b] = wmod[b] @ x[b%2]        (256x1024, K=256)
    wmma_gemm_kernel<0, 0, 4, 0><<<dim3(16, 16, 8), 32, 0, stream>>>(
        wmod, 65536LL, 7u, 256,
        x,    262144LL, 1u, 1024,
        nullptr,
        qt,   262144LL, 1024, 256);

    offsets_kernel<<<256, 256, 0, stream>>>(qt, dw_w, dw_b, ln_w, ln_b, off_w, pos_out, ref_out);
    gridsample_kernel<<<256, 256, 0, stream>>>(x, pos_out, xs);

    // kT[b] = (Wk @ xs[b] + bk)^T  stored (ns, C) ; v[b] = Wv @ xs[b] + bv  stored (C, ns)
    wmma_gemm_kernel<0, 1, 4, 1><<<dim3(16, 16, 8), 32, 0, stream>>>(
        Wk, 0LL, 0u, 256,
        xs, 262144LL, 7u, 1024,
        bk,
        kbufT, 262144LL, 256, 256);
    wmma_gemm_kernel<0, 1, 4, 0><<<dim3(16, 16, 8), 32, 0, stream>>>(
        Wv, 0LL, 0u, 256,
        xs, 262144LL, 7u, 1024,
        bv,
        vbuf, 262144LL, 1024, 256);

    attn_kernel<<<dim3(128, 64), 32, 0, stream>>>(q, kbufT, vbuf, attnout);

    // y = attnout @ Wo^T + bo         (16384x256, K=256)
    wmma_gemm_kernel<1, 2, 4, 0><<<dim3(4, 1024, 1), 32, 0, stream>>>(
        attnout, 0LL, 0u, 256,
        Wo, 0LL, 0u, 256,
        bo,
        y_out, 0LL, 256, 256);
}